// TinyRecurrentSpikingBlock_27616639713853
// MI455X (gfx1250) — compile-verified
//
#include <hip/hip_runtime.h>

#define T_STEPS 512
#define BATCH   128
#define D_INP   256
#define D_HID   512
#define D_OUTP  256
#define BETA    0.9f
#define THRESH  1.0f

typedef float v2f __attribute__((ext_vector_type(2)));
typedef float v8f __attribute__((ext_vector_type(8)));

// ---------------------------------------------------------------------------
// FP32 WMMA GEMM:  C[M,N] = A[M,K] @ B[K,N] + bias[N]
// wave  : 64(M) x 32(N) tile, 4x2 accumulators, K-loop in steps of 4
//         -> 8 v_wmma per 8 loads (4 A b64 + 4 B b32)
// block : 8 waves -> 512 rows of M
// grid  : (N/32, M/512)  -- N-fastest so A rows stay hot in L2 across N tiles
// A frag (16x4 f32, 2 VGPR): lane L -> row = L&15, cols = k + 2*(L>>4) + {0,1}
// B frag (4x16 f32, 2 VGPR): lane L -> col = L&15, rows = k + 2*(L>>4) + {0,1}
// C/D    (16x16 f32, 8 VGPR): elem r, lane L -> row = r + 8*(L>>4), col = L&15
// ---------------------------------------------------------------------------
template<int K, int N>
__global__ __launch_bounds__(256)
void wmma_gemm_f32(const float* __restrict__ A,
                   const float* __restrict__ Bm,
                   const float* __restrict__ bias,
                   float* __restrict__ C) {
  const int lane = threadIdx.x & 31;
  const int wave = threadIdx.x >> 5;
  const int l16  = lane & 15;
  const int half = lane >> 4;           // 0 or 1

  const int n0 = blockIdx.x * 32;
  const int m0 = blockIdx.y * 512 + wave * 64;

  v8f acc[4][2];
#pragma unroll
  for (int mt = 0; mt < 4; ++mt)
#pragma unroll
    for (int nt = 0; nt < 2; ++nt) acc[mt][nt] = v8f{};

  // per-lane base pointers
  const float* aBase = A  + (size_t)(m0 + l16) * K + 2 * half;  // + mt*16*K + k
  const float* bBase = Bm + (size_t)(2 * half) * N + n0 + l16;  // + nt*16 + k*N

#pragma unroll 2
  for (int k = 0; k < K; k += 4) {
    v2f bf[2];
#pragma unroll
    for (int nt = 0; nt < 2; ++nt) {
      bf[nt].x = bBase[(size_t)k * N + nt * 16];
      bf[nt].y = bBase[(size_t)k * N + nt * 16 + N];
    }
#pragma unroll
    for (int mt = 0; mt < 4; ++mt) {
      v2f af = *(const v2f*)(aBase + (size_t)mt * 16 * K + k);
#pragma unroll
      for (int nt = 0; nt < 2; ++nt) {
        acc[mt][nt] = __builtin_amdgcn_wmma_f32_16x16x4_f32(
            /*neg_a=*/false, af, /*neg_b=*/false, bf[nt],
            /*c_mod=*/(short)0, acc[mt][nt],
            /*reuse_a=*/false, /*reuse_b=*/false);
      }
    }
  }

  const float bv0 = bias[n0 + l16];
  const float bv1 = bias[n0 + 16 + l16];
#pragma unroll
  for (int mt = 0; mt < 4; ++mt) {
#pragma unroll
    for (int r = 0; r < 8; ++r) {
      const int row = m0 + mt * 16 + r + 8 * half;
      float* cRow = C + (size_t)row * N + n0 + l16;
      cRow[0]  = acc[mt][0][r] + bv0;
      cRow[16] = acc[mt][1][r] + bv1;
    }
  }
}

// ---------------------------------------------------------------------------
// Elementwise RLIF scan over T. One thread per (b,h) element; reads cur[t],
// overwrites it in place with the binary spike (each element is read exactly
// once before being dead). Deterministic per-block spike-count partials.
// ---------------------------------------------------------------------------
__global__ __launch_bounds__(256)
void rlif_scan(float* __restrict__ curspk,
               const float* __restrict__ w_rec,
               float* __restrict__ partials) {
  const int idx = blockIdx.x * 256 + threadIdx.x;   // b*H + h
  const int h   = idx & (D_HID - 1);
  const float wr = w_rec[h] - THRESH;               // spikes*w_rec - spikes*TH

  float V = 0.0f, cnt = 0.0f;
  const size_t stride = (size_t)BATCH * D_HID;
  size_t off = idx;
  for (int t = 0; t < T_STEPS; ++t, off += stride) {
    const float c = curspk[off];
    const float s = (V > THRESH) ? 1.0f : 0.0f;     // heaviside(V - TH)
    curspk[off] = s;                                // spike for GEMM2 (exact)
    V = BETA * V + c + s * wr;
    cnt += s;
  }

  __shared__ float sdata[256];
  sdata[threadIdx.x] = cnt;
  __syncthreads();
  for (int sft = 128; sft > 0; sft >>= 1) {
    if (threadIdx.x < sft) sdata[threadIdx.x] += sdata[threadIdx.x + sft];
    __syncthreads();
  }
  if (threadIdx.x == 0) partials[blockIdx.x] = sdata[0];
}

// ---------------------------------------------------------------------------
// readout[b,o] = mean_t logits[t,b,o] ; spike_rate = sum(partials)/(T*B*H)
// ---------------------------------------------------------------------------
__global__ __launch_bounds__(256)
void finalize(const float* __restrict__ logits,
              const float* __restrict__ partials,
              float* __restrict__ readout,
              float* __restrict__ spike_rate) {
  const int tid = blockIdx.x * 256 + threadIdx.x;   // < B*O
  float sum = 0.0f;
  const size_t stride = (size_t)BATCH * D_OUTP;
  size_t off = tid;
  for (int t = 0; t < T_STEPS; ++t, off += stride) sum += logits[off];
  readout[tid] = sum * (1.0f / T_STEPS);

  if (tid == 0) {
    float s = 0.0f;
    for (int i = 0; i < (BATCH * D_HID) / 256; ++i) s += partials[i];
    spike_rate[0] = s / ((float)T_STEPS * BATCH * D_HID);
  }
}

// ---------------------------------------------------------------------------
extern "C" void kernel_launch(void* const* d_in, const int* in_sizes, int n_in,
                              void* d_out, int out_size, void* d_ws, size_t ws_size,
                              hipStream_t stream) {
  const float* x_seq  = (const float*)d_in[0];  // [T, B, D_IN]
  const float* W_in   = (const float*)d_in[1];  // [D_IN, D_H]
  const float* b_in   = (const float*)d_in[2];  // [D_H]
  const float* w_rec  = (const float*)d_in[3];  // [D_H]
  const float* W_head = (const float*)d_in[4];  // [D_H, D_OUT]
  const float* b_head = (const float*)d_in[5];  // [D_OUT]

  float* out        = (float*)d_out;
  float* readout    = out;                                          // B*O
  float* logits     = out + (size_t)BATCH * D_OUTP;                 // T*B*O
  float* spike_rate = logits + (size_t)T_STEPS * BATCH * D_OUTP;    // 1

  float* curspk   = (float*)d_ws;                                   // T*B*H f32
  float* partials = curspk + (size_t)T_STEPS * BATCH * D_HID;       // 256 f32

  const int M = T_STEPS * BATCH;  // 65536

  // 1) CUR = X @ W_in + b_in       [65536,256]x[256,512]
  {
    dim3 grid(D_HID / 32, M / 512);
    wmma_gemm_f32<D_INP, D_HID><<<grid, 256, 0, stream>>>(x_seq, W_in, b_in, curspk);
  }

  // 2) RLIF scan: CUR -> SPIKES (in place) + spike-count partials
  rlif_scan<<<(BATCH * D_HID) / 256, 256, 0, stream>>>(curspk, w_rec, partials);

  // 3) LOGITS = SPIKES @ W_head + b_head   [65536,512]x[512,256]
  {
    dim3 grid(D_OUTP / 32, M / 512);
    wmma_gemm_f32<D_HID, D_OUTP><<<grid, 256, 0, stream>>>(curspk, W_head, b_head, logits);
  }

  // 4) readout mean over T + spike rate
  finalize<<<(BATCH * D_OUTP) / 256, 256, 0, stream>>>(logits, partials, readout, spike_rate);
}